// opt_kde_45664092291721
// MI455X (gfx1250) — compile-verified
//
#include <hip/hip_runtime.h>
#include <hip/hip_bf16.h>

typedef __attribute__((ext_vector_type(2))) float v2f;
typedef __attribute__((ext_vector_type(8))) float v8f;

#define E_ 4
#define L_ 10
#define M_ 512
#define F_ 128
#define S_ 1000
#define NPAIR 6
#define LOG2E 1.4426950408889634f

// ---------------------------------------------------------------------------
// Kernel 1: per-(e,l) partial sum of per-feature stds over the M dim.
// grid = E*L blocks, 128 threads (one per feature). Coalesced loads.
// ---------------------------------------------------------------------------
__global__ __launch_bounds__(128) void col_std_partial(
    const float* __restrict__ matrix, float* __restrict__ partial) {
  const int el = blockIdx.x;          // e*L + l
  const int f  = threadIdx.x;
  const float* base = matrix + (size_t)el * M_ * F_ + f;
  float s = 0.f, s2 = 0.f;
  for (int m = 0; m < M_; ++m) {
    float v = base[(size_t)m * F_];
    s += v; s2 += v * v;
  }
  const float mean = s * (1.0f / M_);
  float var = s2 * (1.0f / M_) - mean * mean;
  var = fmaxf(var, 0.0f);
  float sd = sqrtf(var);

  __shared__ float lds[128];
  lds[f] = sd;
  __syncthreads();
  for (int off = 64; off > 0; off >>= 1) {
    if (f < off) lds[f] += lds[f + off];
    __syncthreads();
  }
  if (f == 0) partial[el] = lds[0];
}

// ---------------------------------------------------------------------------
// Kernel 2: reduce 40 partials -> std, bw, exp coefficient, divisor.
// ---------------------------------------------------------------------------
__global__ __launch_bounds__(64) void finalize_std(
    const float* __restrict__ partial, float* __restrict__ consts) {
  __shared__ float lds[64];
  const int t = threadIdx.x;
  lds[t] = (t < E_ * L_) ? partial[t] : 0.0f;
  __syncthreads();
  for (int off = 32; off > 0; off >>= 1) {
    if (t < off) lds[t] += lds[t + off];
    __syncthreads();
  }
  if (t == 0) {
    const float total = lds[0];
    const float nelf  = (float)(E_ * L_ * F_);
    const float stdg  = total / nelf;                  // std of matrix
    const float ratio = total / (nelf * stdg);         // std(mat).mean() (~1.0)
    const float bw    = 1.06f * exp2f(-9.0f / 5.0f) * ratio; // 1.06 * 512^(-1/5)
    consts[0] = 1.0f / stdg;                           // inv_std
    consts[1] = bw;
    consts[2] = (-0.5f / (bw * bw)) * LOG2E;           // offset^d2 = exp2(c2*d2)
    consts[3] = sqrtf(2.0f * 3.14159265358979323846f) * bw; // divisor
  }
}

// ---------------------------------------------------------------------------
// Kernel 3: proj = (matrix * inv_std) @ params via V_WMMA_F32_16X16X4_F32.
// grid.x = (E*L*M)/16 row tiles; 8 waves/block, each wave owns one 16-col tile.
// A 16x4 f32 layout: lanes 0-15 = rows, VGPR0/1 = K0/K1; lanes 16-31 = K2/K3.
// B 4x16 layout: lane n holds B[k,n] striped across 2 VGPRs (K pairs by lane half).
// ---------------------------------------------------------------------------
__global__ __launch_bounds__(256) void proj_gemm_wmma(
    const float* __restrict__ matrix, const float* __restrict__ params,
    const float* __restrict__ consts, float* __restrict__ proj) {
  const float inv_std = consts[0];
  const int lane = threadIdx.x & 31;
  const int wave = threadIdx.x >> 5;
  const int r0 = blockIdx.x * 16;      // row tile over E*L*M rows
  const int g0 = wave * 16;            // col tile
  const int sub = lane & 15;           // A: row-in-tile; B/D: col-in-tile
  const int kh  = lane >> 4;           // K-pair selector (0 or 1)

  const float* arow = matrix + (size_t)(r0 + sub) * F_;
  v8f acc = {};
#pragma unroll
  for (int k = 0; k < F_; k += 4) {
    const int ka = k + 2 * kh;
    v2f a, b;
    a[0] = arow[ka]     * inv_std;
    a[1] = arow[ka + 1] * inv_std;
    b[0] = params[(size_t)ka * F_ + g0 + sub];
    b[1] = params[(size_t)(ka + 1) * F_ + g0 + sub];
    acc = __builtin_amdgcn_wmma_f32_16x16x4_f32(
        false, a, false, b, (short)0, acc, false, false);
  }
#pragma unroll
  for (int j = 0; j < 8; ++j) {
    const int row = r0 + j + kh * 8;   // lanes>=16 hold rows 8..15
    proj[(size_t)row * F_ + g0 + sub] = acc[j];
  }
}

// ---------------------------------------------------------------------------
// Kernel 4a/4b: global min/max over proj (two-stage).
// ---------------------------------------------------------------------------
__global__ __launch_bounds__(256) void minmax_partial(
    const float* __restrict__ proj, float* __restrict__ mn, float* __restrict__ mx) {
  const int N = E_ * L_ * M_ * F_;
  float lo =  __builtin_inff();
  float hi = -__builtin_inff();
  for (int i = blockIdx.x * 256 + threadIdx.x; i < N; i += 256 * 256) {
    const float v = proj[i];
    lo = fminf(lo, v); hi = fmaxf(hi, v);
  }
  __shared__ float slo[256], shi[256];
  const int t = threadIdx.x;
  slo[t] = lo; shi[t] = hi;
  __syncthreads();
  for (int off = 128; off > 0; off >>= 1) {
    if (t < off) {
      slo[t] = fminf(slo[t], slo[t + off]);
      shi[t] = fmaxf(shi[t], shi[t + off]);
    }
    __syncthreads();
  }
  if (t == 0) { mn[blockIdx.x] = slo[0]; mx[blockIdx.x] = shi[0]; }
}

__global__ __launch_bounds__(256) void finalize_grid(
    const float* __restrict__ mn, const float* __restrict__ mx,
    float* __restrict__ consts) {
  __shared__ float slo[256], shi[256];
  const int t = threadIdx.x;
  slo[t] = mn[t]; shi[t] = mx[t];
  __syncthreads();
  for (int off = 128; off > 0; off >>= 1) {
    if (t < off) {
      slo[t] = fminf(slo[t], slo[t + off]);
      shi[t] = fmaxf(shi[t], shi[t + off]);
    }
    __syncthreads();
  }
  if (t == 0) {
    const float left = slo[0], right = shi[0];
    consts[4] = left;
    consts[5] = right;
    consts[6] = (right - left) / (float)S_;        // delta
    consts[7] = (right - left) / (float)(S_ - 1);  // linspace step
  }
}

// ---------------------------------------------------------------------------
// Kernel 5: the KDE pairwise-L1 accumulation. One block per (l,f) column.
// proj[:, l, :, f] (4 envs x 512 samples) staged in LDS; threads stride grid
// points; 2.62e9 exp2 evaluations total across the launch.
// ---------------------------------------------------------------------------
__global__ __launch_bounds__(256) void kde_dist(
    const float* __restrict__ proj, const int* __restrict__ data_len,
    const unsigned char* __restrict__ is_train, const float* __restrict__ consts,
    float* __restrict__ out) {
  __shared__ float vals[E_ * M_];     // 8 KB
  __shared__ float red[NPAIR * 256];  // 6 KB
  const int tid = threadIdx.x;
  const int f = blockIdx.x & (F_ - 1);
  const int l = blockIdx.x >> 7;

  const float k2      = consts[2];
  const float divisor = consts[3];
  const float left    = consts[4];
  const float delta   = consts[6];
  const float step    = consts[7];

  for (int idx = tid; idx < E_ * M_; idx += 256) {
    const int e = idx >> 9, m = idx & (M_ - 1);
    vals[idx] = proj[((size_t)(e * L_ + l) * M_ + m) * F_ + f];
  }
  __syncthreads();

  float lenf[E_], cnt0[E_];
#pragma unroll
  for (int e = 0; e < E_; ++e) {
    lenf[e] = (float)data_len[e * L_ + l];
    cnt0[e] = (float)M_ - lenf[e];
  }

  float pacc[NPAIR] = {0, 0, 0, 0, 0, 0};
  for (int p = tid; p < S_; p += 256) {
    const float x = left + step * (float)p;
    float r[E_];
#pragma unroll
    for (int e = 0; e < E_; ++e) {
      const float* v = &vals[e * M_];
      float s0 = 0.f, s1 = 0.f, s2 = 0.f, s3 = 0.f;
      for (int m = 0; m < M_; m += 4) {
        float d0 = v[m]     - x;
        float d1 = v[m + 1] - x;
        float d2 = v[m + 2] - x;
        float d3 = v[m + 3] - x;
        s0 += __builtin_amdgcn_exp2f(k2 * d0 * d0);
        s1 += __builtin_amdgcn_exp2f(k2 * d1 * d1);
        s2 += __builtin_amdgcn_exp2f(k2 * d2 * d2);
        s3 += __builtin_amdgcn_exp2f(k2 * d3 * d3);
      }
      r[e] = (s0 + s1) + (s2 + s3);
    }
    const float ex = __builtin_amdgcn_exp2f(k2 * x * x);  // offset^(x^2)
#pragma unroll
    for (int e = 0; e < E_; ++e) r[e] = (r[e] - cnt0[e] * ex) / lenf[e];
    int pi = 0;
#pragma unroll
    for (int i = 0; i < E_; ++i)
#pragma unroll
      for (int j = i + 1; j < E_; ++j, ++pi) pacc[pi] += fabsf(r[i] - r[j]);
  }

#pragma unroll
  for (int q = 0; q < NPAIR; ++q) red[q * 256 + tid] = pacc[q];
  __syncthreads();
  for (int off = 128; off > 0; off >>= 1) {
    if (tid < off) {
#pragma unroll
      for (int q = 0; q < NPAIR; ++q)
        red[q * 256 + tid] += red[q * 256 + tid + off];
    }
    __syncthreads();
  }

  if (tid == 0) {
    const float scale = (delta * 0.5f) / divisor;   // (store_dis/divisor)*(delta/2)
    float tests  = -__builtin_inff();
    float trains = -__builtin_inff();
    int pi = 0;
    for (int i = 0; i < E_; ++i)
      for (int j = i + 1; j < E_; ++j, ++pi) {
        const float h = red[pi * 256] * scale;
        tests = fmaxf(tests, h);
        if (is_train[i] && is_train[j]) trains = fmaxf(trains, h);
      }
    out[l * F_ + f]            = trains;            // train_results [L,F]
    out[L_ * F_ + l * F_ + f]  = tests;             // test_results  [L,F]
  }
}

// ---------------------------------------------------------------------------
// Kernel 6: train_dis / test_dis = max over L, mean over F.
// ---------------------------------------------------------------------------
__global__ __launch_bounds__(128) void final_scalars(float* __restrict__ out) {
  __shared__ float a[128], b[128];
  const int f = threadIdx.x;
  float tmax = -__builtin_inff(), smax = -__builtin_inff();
  for (int l = 0; l < L_; ++l) {
    tmax = fmaxf(tmax, out[l * F_ + f]);
    smax = fmaxf(smax, out[L_ * F_ + l * F_ + f]);
  }
  a[f] = tmax; b[f] = smax;
  __syncthreads();
  for (int off = 64; off > 0; off >>= 1) {
    if (f < off) { a[f] += a[f + off]; b[f] += b[f + off]; }
    __syncthreads();
  }
  if (f == 0) {
    out[2 * L_ * F_]     = a[0] * (1.0f / F_);   // train_dis
    out[2 * L_ * F_ + 1] = b[0] * (1.0f / F_);   // test_dis
  }
}

// ---------------------------------------------------------------------------
// Launcher. ws layout (floats): [0..7] consts, [8..47] std partials,
// [64..319] block mins, [320..575] block maxs, [1024..] proj (2,621,440 f32).
// ---------------------------------------------------------------------------
extern "C" void kernel_launch(void* const* d_in, const int* in_sizes, int n_in,
                              void* d_out, int out_size, void* d_ws, size_t ws_size,
                              hipStream_t stream) {
  const float* matrix = (const float*)d_in[0];
  const float* params = (const float*)d_in[1];
  const int* data_len = (const int*)d_in[2];
  const unsigned char* is_train = (const unsigned char*)d_in[3];
  float* out = (float*)d_out;

  float* ws      = (float*)d_ws;
  float* consts  = ws;
  float* partial = ws + 8;
  float* mn      = ws + 64;
  float* mx      = ws + 320;
  float* proj    = ws + 1024;

  col_std_partial<<<E_ * L_, 128, 0, stream>>>(matrix, partial);
  finalize_std<<<1, 64, 0, stream>>>(partial, consts);
  proj_gemm_wmma<<<(E_ * L_ * M_) / 16, 256, 0, stream>>>(matrix, params, consts, proj);
  minmax_partial<<<256, 256, 0, stream>>>(proj, mn, mx);
  finalize_grid<<<1, 256, 0, stream>>>(mn, mx, consts);
  kde_dist<<<L_ * F_, 256, 0, stream>>>(proj, data_len, is_train, consts, out);
  final_scalars<<<1, 128, 0, stream>>>(out);
}